// LightGCNLayer_32658931319095
// MI455X (gfx1250) — compile-verified
//
#include <hip/hip_runtime.h>
#include <hip/hip_bf16.h>

#ifndef D_FEAT
#define D_FEAT 64
#endif
#define STAGES 4            // async pipeline depth (power of two)
#define WAVES_PER_BLOCK 8   // 256 threads / wave32

typedef __attribute__((address_space(3))) float lds_float_t;

// Raw LDS byte offset of a __shared__ pointer (AS3 pointer value == LDS addr).
__device__ __forceinline__ unsigned lds_byte_offset(float* p) {
    return (unsigned)(size_t)(lds_float_t*)p;
}

// ---------------------------------------------------------------------------
// Kernel 1: zero the output [N,64] (float4 stores) and the degree scratch [N]
// ---------------------------------------------------------------------------
__global__ void lgcn_init(float4* __restrict__ out4, float* __restrict__ deg,
                          int n_out4, int n_nodes) {
    const int tid    = blockIdx.x * blockDim.x + threadIdx.x;
    const int stride = gridDim.x * blockDim.x;
    const float4 z = make_float4(0.f, 0.f, 0.f, 0.f);
    for (int i = tid; i < n_out4; i += stride) out4[i] = z;
    for (int i = tid; i < n_nodes; i += stride) deg[i] = 0.f;
}

// ---------------------------------------------------------------------------
// Kernel 2: degree of each node as a source (bincount over row)
// ---------------------------------------------------------------------------
__global__ void lgcn_degree(const int* __restrict__ row,
                            float* __restrict__ deg, int n_edges) {
    const int e = blockIdx.x * blockDim.x + threadIdx.x;
    if (e < n_edges) {
#if defined(__HIP_DEVICE_COMPILE__)
        unsafeAtomicAdd(&deg[row[e]], 1.0f);   // native global_atomic_add_f32
#else
        atomicAdd(&deg[row[e]], 1.0f);
#endif
    }
}

// ---------------------------------------------------------------------------
// Kernel 3: deg -> deg^{-1/2} (0 where deg==0), in place
// ---------------------------------------------------------------------------
__global__ void lgcn_dis(float* __restrict__ deg, int n_nodes) {
    const int i = blockIdx.x * blockDim.x + threadIdx.x;
    if (i < n_nodes) {
        const float d = deg[i];
        deg[i] = (d > 0.0f) ? __frsqrt_rn(d) : 0.0f;
    }
}

// ---------------------------------------------------------------------------
// Kernel 4: gather-scale-scatter with a 4-deep ASYNC-to-LDS gather pipeline.
//   Each wave32 covers 2 edges/iter: lanes 0-15 -> edge A, lanes 16-31 ->
//   edge B; each lane owns a float4 feature slice (16 lanes x 4 = 64 feats).
//   The x[col] gather is staged into LDS with global_load_async_to_lds_b128
//   (ASYNCcnt-tracked, zero VGPR cost), 4 stages in flight per wave; the
//   consume side waits with s_wait_asynccnt 3, reads the staged float4 from
//   LDS, scales by dis[row]*dis[col], and scatters with native f32 atomics.
// ---------------------------------------------------------------------------
__global__ void __launch_bounds__(256)
lgcn_scatter(const float* __restrict__ x,
             const int*   __restrict__ row,
             const int*   __restrict__ col,
             const float* __restrict__ dis,
             float*       __restrict__ out,
             int n_edges) {
    // [wave][stage][2 edges * 64 feats] = 8*4*128*4B = 16 KB / block
    __shared__ float smem[WAVES_PER_BLOCK][STAGES][2 * D_FEAT];

    const int lane    = threadIdx.x & 31;
    const int waveInB = threadIdx.x >> 5;
    const int half    = lane >> 4;           // which edge of the pair
    const int feat    = (lane & 15) << 2;    // float4 slice within D=64
    const int waveId  = (blockIdx.x * blockDim.x + threadIdx.x) >> 5;
    const int nWaves  = (gridDim.x * blockDim.x) >> 5;
    const int step    = nWaves * 2;
    const int start   = waveId * 2;
    const int slot    = half * D_FEAT + feat;

    const int nIter = (start < n_edges) ? ((n_edges - 1 - start) / step + 1) : 0;

    // Issue one async 16B/lane gather for pipeline stage s, iteration base.
    // Tail-clamped so every slot always issues exactly one op (uniform
    // ASYNCcnt bookkeeping); garbage stages are never consumed.
    auto issue_stage = [&](int s, int ebase) {
        int e = ebase + half;
        e = (e < n_edges) ? e : (n_edges - 1);
        const int c = col[e];
        const float*   gp  = x + (size_t)c * D_FEAT + feat;
        const unsigned lds = lds_byte_offset(&smem[waveInB][s][slot]);
        asm volatile("global_load_async_to_lds_b128 %0, %1, off"
                     :: "v"(lds), "v"(gp)
                     : "memory");
    };

    // Prologue: fill the pipeline.
    for (int s = 0; s < STAGES; ++s) issue_stage(s, start + s * step);

    for (int i = 0; i < nIter; ++i) {
        const int base = start + i * step;
        const int s    = i & (STAGES - 1);

        // Oldest outstanding stage (this one) has landed in LDS.
        asm volatile("s_wait_asynccnt 3" ::: "memory");

        const float4 v =
            *reinterpret_cast<const float4*>(&smem[waveInB][s][slot]);

        const int e = base + half;
        if (e < n_edges) {
            const int r = row[e];
            const int c = col[e];
            const float norm = dis[r] * dis[c];
            float* o = out + (size_t)r * D_FEAT + feat;
#if defined(__HIP_DEVICE_COMPILE__)
            unsafeAtomicAdd(o + 0, v.x * norm);
            unsafeAtomicAdd(o + 1, v.y * norm);
            unsafeAtomicAdd(o + 2, v.z * norm);
            unsafeAtomicAdd(o + 3, v.w * norm);
#else
            atomicAdd(o + 0, v.x * norm);
            atomicAdd(o + 1, v.y * norm);
            atomicAdd(o + 2, v.z * norm);
            atomicAdd(o + 3, v.w * norm);
#endif
        }

        // Close the DS-read vs ASYNC-write WAR hazard explicitly: async LDS
        // writes are unordered w.r.t. DScnt-tracked reads, so drain the
        // consume ds_load before reusing this stage's LDS slot.
        asm volatile("s_wait_dscnt 0x0" ::: "memory");

        // Refill this stage for iteration i+STAGES.
        issue_stage(s, start + (i + STAGES) * step);
    }
    // Leftover clamped async ops drain at s_endpgm (implicit wait-idle).
}

// ---------------------------------------------------------------------------
extern "C" void kernel_launch(void* const* d_in, const int* in_sizes, int n_in,
                              void* d_out, int out_size, void* d_ws, size_t ws_size,
                              hipStream_t stream) {
    const float* x       = (const float*)d_in[0];
    const int*   eidx    = (const int*)d_in[1];     // [2, E] row-major int32
    const int    n_nodes = in_sizes[0] / D_FEAT;    // 100000
    const int    n_edges = in_sizes[1] / 2;         // 1600000
    const int*   row     = eidx;                    // edge_index[0]
    const int*   col     = eidx + n_edges;          // edge_index[1]

    float* out = (float*)d_out;                     // [N, 64] f32
    float* deg = (float*)d_ws;                      // N floats scratch

    // 1) zero out + deg
    {
        const int n_out4 = out_size / 4;
        lgcn_init<<<1024, 256, 0, stream>>>(
            reinterpret_cast<float4*>(out), deg, n_out4, n_nodes);
    }
    // 2) degrees
    {
        const int blocks = (n_edges + 255) / 256;
        lgcn_degree<<<blocks, 256, 0, stream>>>(row, deg, n_edges);
    }
    // 3) deg^{-1/2}
    {
        const int blocks = (n_nodes + 255) / 256;
        lgcn_dis<<<blocks, 256, 0, stream>>>(deg, n_nodes);
    }
    // 4) async-pipelined gather-scale-scatter
    {
        // 2048 blocks * 8 waves = 16384 waves, 2 edges/wave/iter
        // -> ~49 pipelined iterations per wave.
        lgcn_scatter<<<2048, 256, 0, stream>>>(x, row, col, deg, out, n_edges);
    }
}